// MPSTPCP_8435315769675
// MI455X (gfx1250) — compile-verified
//
#include <hip/hip_runtime.h>

typedef __attribute__((ext_vector_type(2))) float v2f;
typedef __attribute__((ext_vector_type(8))) float v8f;

#define L_LAYERS 783
#define B_BATCH  8192
#define N_SITES  784

// ---------------------------------------------------------------------------
// Kernel 1: build per-layer real superoperators T_l (4 outputs x 12 inputs).
// Stored in WMMA-A-operand-friendly layout:
//   TA[l*48 + t*16 + i*4 + k] = T_l[i][4*t + k]
// where i = output DOF (s00', s11', Re s01', Im s01'),
//       t = x-monomial (x0^2, x0*x1, x1^2), k = input sigma DOF.
// One thread per (layer, probe-column): probe the channel with basis inputs.
// ---------------------------------------------------------------------------
__global__ void build_superops(const float* __restrict__ kre,
                               const float* __restrict__ kim,
                               float* __restrict__ TA) {
  int idx = blockIdx.x * blockDim.x + threadIdx.x;
  if (idx >= L_LAYERS * 12) return;
  int l = idx / 12;
  int e = idx % 12;
  int t = e >> 2;   // monomial index
  int j = e & 3;    // sigma basis index

  // sigma basis (2x2 complex, Hermitian)
  float sr[2][2] = {{0.f,0.f},{0.f,0.f}}, si[2][2] = {{0.f,0.f},{0.f,0.f}};
  if (j == 0)      sr[0][0] = 1.f;
  else if (j == 1) sr[1][1] = 1.f;
  else if (j == 2) { sr[0][1] = 1.f; sr[1][0] = 1.f; }
  else             { si[0][1] = 1.f; si[1][0] = -1.f; }
  // x-outer basis (2x2 real symmetric)
  float xm[2][2] = {{0.f,0.f},{0.f,0.f}};
  if (t == 0)      xm[0][0] = 1.f;
  else if (t == 1) { xm[0][1] = 1.f; xm[1][0] = 1.f; }
  else             xm[1][1] = 1.f;

  // rho[(2p+q)][(2r+s)] = sigma[p][r] * xm[q][s]   (4x4 complex)
  float rr[4][4], ri[4][4];
#pragma unroll
  for (int p = 0; p < 2; ++p)
#pragma unroll
    for (int q = 0; q < 2; ++q)
#pragma unroll
      for (int r = 0; r < 2; ++r)
#pragma unroll
        for (int s = 0; s < 2; ++s) {
          rr[2*p+q][2*r+s] = sr[p][r] * xm[q][s];
          ri[2*p+q][2*r+s] = si[p][r] * xm[q][s];
        }

  // chi = sum_k K_k rho K_k^dagger
  float cr[4][4], ci[4][4];
#pragma unroll
  for (int a = 0; a < 4; ++a)
#pragma unroll
    for (int d = 0; d < 4; ++d) { cr[a][d] = 0.f; ci[a][d] = 0.f; }

  for (int k = 0; k < 16; ++k) {
    const float* Kr = kre + (size_t)(l * 16 + k) * 16;
    const float* Ki = kim + (size_t)(l * 16 + k) * 16;
    float tr[4][4], ti[4][4];
#pragma unroll
    for (int a = 0; a < 4; ++a)
#pragma unroll
      for (int c = 0; c < 4; ++c) {
        float s0 = 0.f, s1 = 0.f;
#pragma unroll
        for (int b = 0; b < 4; ++b) {
          float wr = Kr[a*4+b], wi = Ki[a*4+b];
          s0 += wr * rr[b][c] - wi * ri[b][c];
          s1 += wr * ri[b][c] + wi * rr[b][c];
        }
        tr[a][c] = s0; ti[a][c] = s1;
      }
#pragma unroll
    for (int a = 0; a < 4; ++a)
#pragma unroll
      for (int d = 0; d < 4; ++d) {
        float s0 = 0.f, s1 = 0.f;
#pragma unroll
        for (int c = 0; c < 4; ++c) {
          float wr = Kr[d*4+c], wi = Ki[d*4+c];
          // tmp[a][c] * conj(K[d][c])
          s0 += tr[a][c] * wr + ti[a][c] * wi;
          s1 += ti[a][c] * wr - tr[a][c] * wi;
        }
        cr[a][d] += s0; ci[a][d] += s1;
      }
  }

  // partial trace over first qubit: sigma'[m][n] = chi[m][n] + chi[2+m][2+n]
  float s00r = cr[0][0] + cr[2][2];
  float s11r = cr[1][1] + cr[3][3];
  float s01r = cr[0][1] + cr[2][3];
  float s01i = ci[0][1] + ci[2][3];

  float* base = TA + (size_t)l * 48 + t * 16 + j;   // column (t, k=j)
  base[0]  = s00r;  // output i=0
  base[4]  = s11r;  // output i=1
  base[8]  = s01r;  // output i=2
  base[12] = s01i;  // output i=3
}

// ---------------------------------------------------------------------------
// Kernel 2: batched scan. One wave = 16 batch elements (batch = lane & 15).
// Per step: 3x V_WMMA_F32_16X16X4_F32 apply the 4x12 superop to 16 batches.
// A = T_l with rows replicated T[M & 3]  ->  C VGPR i holds u'_i for
// batch (lane & 15) on ALL 32 lanes (row i+8 duplicates row i), so the
// recurrence needs no cross-lane data movement.
// Normalization uses hardware v_rcp_f32 (approx, ~1 ulp) instead of the
// IEEE div expansion (v_div_scale/v_div_fmas/v_div_fixup): ~7 fewer VALU
// ops per step in a loop whose body is only ~25 VALU ops.
// ---------------------------------------------------------------------------
__global__ void mps_scan(const float* __restrict__ X,
                         const float* __restrict__ TA,
                         float* __restrict__ out) {
  const int lane = threadIdx.x & 31;
  const int wave = threadIdx.x >> 5;
  const int wavesPerBlock = blockDim.x >> 5;
  const int batch = (blockIdx.x * wavesPerBlock + wave) * 16 + (lane & 15);
  if (batch >= B_BATCH) return;

  const float* Xrow = X + (size_t)batch * (N_SITES * 2);

  // initial state from site 0: sigma = x x^T / |x|^2 (all lanes mirrored)
  float x0 = Xrow[0], x1 = Xrow[1];
  float inv = __builtin_amdgcn_rcpf(x0 * x0 + x1 * x1);
  float u0 = x0 * x0 * inv;   // s00
  float u1 = x1 * x1 * inv;   // s11
  float u2 = x0 * x1 * inv;   // Re s01
  float u3 = 0.0f;            // Im s01

  // per-lane A-operand base: A_t vgpr pair = T_l[lane&3][4t + kh + {0,1}]
  const int kh = (lane >> 4) * 2;                   // 0 (lanes 0-15) or 2
  const float* tbase = TA + (lane & 3) * 4 + kh;

  const v8f zero = {0.f,0.f,0.f,0.f,0.f,0.f,0.f,0.f};

  for (int n = 1; n <= L_LAYERS; ++n) {
    const int l = n - 1;
    // next input pair (normalized monomials via hw reciprocal)
    float2 xv = *(const float2*)(Xrow + n * 2);
    float inv2 = __builtin_amdgcn_rcpf(xv.x * xv.x + xv.y * xv.y);
    float y0 = xv.x * xv.x * inv2;
    float y1 = xv.x * xv.y * inv2;
    float y2 = xv.y * xv.y * inv2;

    // B operand: lanes 0-15 carry w[4t+0]/w[4t+1], lanes 16-31 w[4t+2]/w[4t+3]
    float s02 = (lane < 16) ? u0 : u2;
    float s13 = (lane < 16) ? u1 : u3;
    v2f b0 = {y0 * s02, y0 * s13};
    v2f b1 = {y1 * s02, y1 * s13};
    v2f b2 = {y2 * s02, y2 * s13};

    // A operand: layer-l superop chunks (shared by all waves -> L2 resident)
    const float* tb = tbase + (size_t)l * 48;
    v2f a0 = *(const v2f*)(tb);
    v2f a1 = *(const v2f*)(tb + 16);
    v2f a2 = *(const v2f*)(tb + 32);

    v8f acc;
    acc = __builtin_amdgcn_wmma_f32_16x16x4_f32(false, a0, false, b0,
                                                (short)0, zero, false, false);
    acc = __builtin_amdgcn_wmma_f32_16x16x4_f32(false, a1, false, b1,
                                                (short)0, acc, false, false);
    acc = __builtin_amdgcn_wmma_f32_16x16x4_f32(false, a2, false, b2,
                                                (short)0, acc, false, false);

    u0 = acc[0]; u1 = acc[1]; u2 = acc[2]; u3 = acc[3];
  }

  // final functional = Re sigma'[0,0] after last layer
  if (lane < 16) out[batch] = u0;
}

// ---------------------------------------------------------------------------
extern "C" void kernel_launch(void* const* d_in, const int* in_sizes, int n_in,
                              void* d_out, int out_size, void* d_ws, size_t ws_size,
                              hipStream_t stream) {
  const float* X   = (const float*)d_in[0];  // (8192, 784, 2) f32
  const float* kre = (const float*)d_in[1];  // (783, 16, 4, 4) f32
  const float* kim = (const float*)d_in[2];  // (783, 16, 4, 4) f32
  float* out = (float*)d_out;                // (8192,) f32
  float* TA  = (float*)d_ws;                 // 783*48 floats = 150,336 B

  {
    int total = L_LAYERS * 12;
    int blk = 256;
    int grid = (total + blk - 1) / blk;
    build_superops<<<grid, blk, 0, stream>>>(kre, kim, TA);
  }
  {
    int blk = 128;                            // 4 waves * 16 batches = 64/block
    int grid = B_BATCH / 64;                  // 128 blocks
    mps_scan<<<grid, blk, 0, stream>>>(X, TA, out);
  }
}